// GNN_29867202576676
// MI455X (gfx1250) — compile-verified
//
#include <hip/hip_runtime.h>

#define NOP 50000
#define NM  10000
#define HID 128
#define LDSTR 132   // 128 + 4 padding: conflict-free column reads from LDS

typedef float v2f __attribute__((ext_vector_type(2)));
typedef float v8f __attribute__((ext_vector_type(8)));

__device__ __forceinline__ void atomicAddF(float* p, float v) {
  unsafeAtomicAdd(p, v);   // native global_atomic_add_f32 on gfx1250
}

// ---------------- elementwise helpers ----------------

// init op features (padded to 128 cols) and fan out into both agg buffers
__global__ void k_init_op_fan(const float* __restrict__ x_op, float* __restrict__ xop,
                              float* __restrict__ a0, float* __restrict__ a1, int n) {
  int i = blockIdx.x * blockDim.x + threadIdx.x;
  if (i >= n * HID) return;
  int r = i >> 7, c = i & 127;
  float v = (c < 7) ? x_op[r * 7 + c] : 0.f;
  xop[i] = v; a0[i] = v; a1[i] = v;
}

// init m features: x_m @ Wm_in + bm_in (3->7, padded), fan out into agg buffers
__global__ void k_init_m_fan(const float* __restrict__ x_m, const float* __restrict__ Wm,
                             const float* __restrict__ bm, float* __restrict__ xm,
                             float* __restrict__ a0, float* __restrict__ a1, int n) {
  int i = blockIdx.x * blockDim.x + threadIdx.x;
  if (i >= n * HID) return;
  int r = i >> 7, c = i & 127;
  float s = 0.f;
  if (c < 7) {
    s = bm[c];
    #pragma unroll
    for (int k = 0; k < 3; ++k) s += x_m[r * 3 + k] * Wm[k * 7 + c];
  }
  xm[i] = s; a0[i] = s; a1[i] = s;
}

// relu + fan out: next-layer features AND the GIN self-term of both agg buffers
__global__ void k_relu_fan(const float* __restrict__ s, float* __restrict__ d,
                           float* __restrict__ a0, float* __restrict__ a1, size_t n) {
  size_t i = (size_t)blockIdx.x * blockDim.x + threadIdx.x;
  if (i < n) {
    float v = fmaxf(s[i], 0.f);
    d[i] = v; a0[i] = v; a1[i] = v;
  }
}

__global__ void k_relu_copy(const float* __restrict__ s, float* __restrict__ d, size_t n) {
  size_t i = (size_t)blockIdx.x * blockDim.x + threadIdx.x;
  if (i < n) d[i] = fmaxf(s[i], 0.f);
}

__global__ void k_zero(float* p, int n) {
  int i = blockIdx.x * blockDim.x + threadIdx.x;
  if (i < n) p[i] = 0.f;
}

// ---------------- edge scatter-add (segment_sum) ----------------

// layer 1: 8 valid feature columns, thread per (edge, col)
__global__ void k_scatter8(const float* __restrict__ xs, const int* __restrict__ src,
                           const int* __restrict__ dst, float* __restrict__ agg, int E) {
  long long i = (long long)blockIdx.x * blockDim.x + threadIdx.x;
  if (i >= (long long)E * 8) return;
  int e = (int)(i >> 3);
  int f = (int)i & 7;
  atomicAddF(&agg[(size_t)dst[e] * HID + f], xs[(size_t)src[e] * HID + f]);
}

// layers 2/3: 128 cols -> one wave per edge, edge index is wave-uniform
// (E*32 is a multiple of 32, so every wave is fully active -> readfirstlane safe)
__global__ void k_scatter128(const float* __restrict__ xs, const int* __restrict__ src,
                             const int* __restrict__ dst, float* __restrict__ agg, int E) {
  long long i = (long long)blockIdx.x * blockDim.x + threadIdx.x;
  if (i >= (long long)E * 32) return;
  int e = __builtin_amdgcn_readfirstlane((int)(i >> 5));  // scalar edge id
  int f = ((int)i & 31) * 4;
  const int s = src[e], d = dst[e];                       // s_load'ed
  const float4 v = *(const float4*)(xs + (size_t)s * HID + f);
  float* p = agg + (size_t)d * HID + f;
  atomicAddF(p + 0, v.x);
  atomicAddF(p + 1, v.y);
  atomicAddF(p + 2, v.z);
  atomicAddF(p + 3, v.w);
}

// ---------------- fused GIN MLP ----------------
// out = [acc +=] ( relu( bn( X@W1 + b1 ) ) @ W2 + b2 ),  X: [N,128] padded (valid cols < DIN)
// block: 256 threads = 8 waves, 32 rows x 128 cols of output.
// wave w: row-tile (w&1), col slab [(w>>1)*32, +32)  -> two 16x16 WMMA tiles.
template <int DIN>
__global__ __launch_bounds__(256) void k_mlp(
    const float* __restrict__ X, int N,
    const float* __restrict__ W1, const float* __restrict__ b1,
    const float* __restrict__ g,  const float* __restrict__ be,
    const float* __restrict__ mu, const float* __restrict__ vv,
    const float* __restrict__ W2, const float* __restrict__ b2,
    float* __restrict__ out, int accumulate)
{
  constexpr int KP = (DIN + 3) & ~3;    // 8 (layer 1) or 128

  __shared__ float Xs[32 * LDSTR];
  __shared__ float Hs[32 * LDSTR];

  const int tid  = threadIdx.x;
  const int lane = tid & 31;
  const int wid  = tid >> 5;
  const int half = lane >> 4;   // 0: lanes 0-15, 1: lanes 16-31
  const int ln   = lane & 15;
  const int rt   = wid & 1;             // row tile in block
  const int cbase = (wid >> 1) * 32;    // col slab base

  const int r0 = blockIdx.x * 32;

  // stage X tile (coalesced); KP is a compile-time power of two
  for (int i = tid; i < 32 * KP; i += 256) {
    int r = i / KP;
    int c = i & (KP - 1);
    Xs[r * LDSTR + c] = (r0 + r < N) ? X[(size_t)(r0 + r) * HID + c] : 0.f;
  }

  const int arow = rt * 16 + ln;
  const int c0 = cbase + ln;
  const int c1 = c0 + 16;

  v8f acc0 = {0.f,0.f,0.f,0.f,0.f,0.f,0.f,0.f};
  v8f acc1 = {0.f,0.f,0.f,0.f,0.f,0.f,0.f,0.f};

  // ---- GEMM1: X(16x4 tiles) @ W1 ----
  if constexpr (DIN == HID) {
    __syncthreads();
    #pragma unroll 4
    for (int k = 0; k < KP; k += 4) {
      // A 16x4: lanes 0-15 hold K=k,k+1; lanes 16-31 hold K=k+2,k+3 (ISA layout)
      v2f a;
      a.x = Xs[arow * LDSTR + k + 2 * half];
      a.y = Xs[arow * LDSTR + k + 2 * half + 1];
      int kr0 = k + 2 * half;
      v2f bA, bB;
      bA.x = W1[kr0 * HID + c0];
      bA.y = W1[(kr0 + 1) * HID + c0];
      bB.x = W1[kr0 * HID + c1];
      bB.y = W1[(kr0 + 1) * HID + c1];
      acc0 = __builtin_amdgcn_wmma_f32_16x16x4_f32(false, a, false, bA, (short)0, acc0, false, false);
      acc1 = __builtin_amdgcn_wmma_f32_16x16x4_f32(false, a, false, bB, (short)0, acc1, false, false);
    }
  } else {
    // small DIN: stage W1 zero-padded into LDS, unguarded inner loop
    __shared__ float W1s[KP * HID];
    for (int i = tid; i < KP * HID; i += 256) {
      int r = i >> 7, c = i & 127;
      W1s[i] = (r < DIN) ? W1[r * HID + c] : 0.f;
    }
    __syncthreads();
    #pragma unroll
    for (int k = 0; k < KP; k += 4) {
      v2f a;
      a.x = Xs[arow * LDSTR + k + 2 * half];
      a.y = Xs[arow * LDSTR + k + 2 * half + 1];
      int kr0 = k + 2 * half;
      v2f bA, bB;
      bA.x = W1s[kr0 * HID + c0];
      bA.y = W1s[(kr0 + 1) * HID + c0];
      bB.x = W1s[kr0 * HID + c1];
      bB.y = W1s[(kr0 + 1) * HID + c1];
      acc0 = __builtin_amdgcn_wmma_f32_16x16x4_f32(false, a, false, bA, (short)0, acc0, false, false);
      acc1 = __builtin_amdgcn_wmma_f32_16x16x4_f32(false, a, false, bB, (short)0, acc1, false, false);
    }
  }

  // ---- bias + BatchNorm(eval) + ReLU -> Hs ----
  {
    float B0 = b1[c0], S0 = g[c0] * rsqrtf(vv[c0] + 1e-5f), M0v = mu[c0], A0 = be[c0];
    float B1 = b1[c1], S1 = g[c1] * rsqrtf(vv[c1] + 1e-5f), M1v = mu[c1], A1 = be[c1];
    #pragma unroll
    for (int j = 0; j < 8; ++j) {
      int row = rt * 16 + j + 8 * half;   // C/D layout: vgpr j -> M = j + 8*half
      float h0 = (acc0[j] + B0 - M0v) * S0 + A0;
      float h1 = (acc1[j] + B1 - M1v) * S1 + A1;
      Hs[row * LDSTR + c0] = fmaxf(h0, 0.f);
      Hs[row * LDSTR + c1] = fmaxf(h1, 0.f);
    }
  }
  __syncthreads();

  // ---- GEMM2: H @ W2 (K = 128) ----
  v8f d0 = {0.f,0.f,0.f,0.f,0.f,0.f,0.f,0.f};
  v8f d1 = {0.f,0.f,0.f,0.f,0.f,0.f,0.f,0.f};
  #pragma unroll 4
  for (int k = 0; k < HID; k += 4) {
    v2f a;
    a.x = Hs[arow * LDSTR + k + 2 * half];
    a.y = Hs[arow * LDSTR + k + 2 * half + 1];
    int kr0 = k + 2 * half;
    v2f bA, bB;
    bA.x = W2[kr0 * HID + c0];
    bA.y = W2[(kr0 + 1) * HID + c0];
    bB.x = W2[kr0 * HID + c1];
    bB.y = W2[(kr0 + 1) * HID + c1];
    d0 = __builtin_amdgcn_wmma_f32_16x16x4_f32(false, a, false, bA, (short)0, d0, false, false);
    d1 = __builtin_amdgcn_wmma_f32_16x16x4_f32(false, a, false, bB, (short)0, d1, false, false);
  }

  // ---- + b2, optional accumulate (uniform branch), store ----
  {
    float B0 = b2[c0], B1 = b2[c1];
    if (accumulate) {
      #pragma unroll
      for (int j = 0; j < 8; ++j) {
        int row = r0 + rt * 16 + j + 8 * half;
        if (row < N) {
          size_t o0 = (size_t)row * HID + c0;
          size_t o1 = (size_t)row * HID + c1;
          out[o0] += d0[j] + B0;
          out[o1] += d1[j] + B1;
        }
      }
    } else {
      #pragma unroll
      for (int j = 0; j < 8; ++j) {
        int row = r0 + rt * 16 + j + 8 * half;
        if (row < N) {
          size_t o0 = (size_t)row * HID + c0;
          size_t o1 = (size_t)row * HID + c1;
          out[o0] = d0[j] + B0;
          out[o1] = d1[j] + B1;
        }
      }
    }
  }
}

// ---------------- final pooling: colsum then tiny projection ----------------
__global__ void k_colsum(const float* __restrict__ X, int N, float* __restrict__ colsum) {
  int c = threadIdx.x;             // 128 threads
  int rs = blockIdx.x * 256;
  int re = min(rs + 256, N);
  float s = 0.f;
  for (int r = rs; r < re; ++r) s += X[(size_t)r * HID + c];
  atomicAddF(&colsum[c], s);
}

__global__ void k_final(const float* __restrict__ colsum,
                        const float* __restrict__ Wop, const float* __restrict__ bop,
                        const float* __restrict__ Wmo, const float* __restrict__ bmo,
                        float* __restrict__ out) {
  int j = threadIdx.x;
  if (j < 16) {
    float s = bop[j] * (float)NOP;
    for (int k = 0; k < HID; ++k) s += colsum[k] * Wop[k * 16 + j];
    out[j] = s;
  } else if (j < 32) {
    int jj = j - 16;
    float s = bmo[jj] * (float)NM;
    for (int k = 0; k < HID; ++k) s += colsum[128 + k] * Wmo[k * 16 + jj];
    out[j] = s;
  }
}

// ---------------- host ----------------
extern "C" void kernel_launch(void* const* d_in, const int* in_sizes, int n_in,
                              void* d_out, int out_size, void* d_ws, size_t ws_size,
                              hipStream_t stream) {
  const float* x_op   = (const float*)d_in[0];
  const float* x_m    = (const float*)d_in[1];
  const float* Wm_in  = (const float*)d_in[2];
  const float* bm_in  = (const float*)d_in[3];
  // params.layers: idx = 4 + L*32 + R*8 + P ; R in (oo=0, om=1, mo=2, mm=3),
  // P in (W1,b1,g,be,mu,v,W2,b2)
  const float* Wop    = (const float*)d_in[100];
  const float* bop    = (const float*)d_in[101];
  const float* Wm_out = (const float*)d_in[102];
  const float* bm_out = (const float*)d_in[103];
  const int* src_oo = (const int*)d_in[104];
  const int* dst_oo = (const int*)d_in[105];
  const int* src_om = (const int*)d_in[106];
  const int* dst_om = (const int*)d_in[107];
  const int* src_mo = (const int*)d_in[108];
  const int* dst_mo = (const int*)d_in[109];
  const int* src_mm = (const int*)d_in[110];
  const int* dst_mm = (const int*)d_in[111];
  const int E_oo = in_sizes[104], E_om = in_sizes[106];
  const int E_mo = in_sizes[108], E_mm = in_sizes[110];

  const size_t opF = (size_t)NOP * HID, mF = (size_t)NM * HID;
  float* ws      = (float*)d_ws;
  float* xop     = ws;
  float* xm      = xop + opF;
  float* aggA_op = xm + mF;
  float* aggB_op = aggA_op + opF;
  float* aggA_m  = aggB_op + opF;
  float* aggB_m  = aggA_m + mF;
  float* out_op  = aggB_m + mF;
  float* out_m   = out_op + opF;
  float* colsum  = out_m + mF;          // 256 floats

  auto P = [&](int idx) { return (const float*)d_in[idx]; };
  auto mlp7 = [&](const float* X, int N, int pb, float* out, int acc) {
    k_mlp<7><<<dim3((N + 31) / 32), dim3(256), 0, stream>>>(
        X, N, P(pb), P(pb + 1), P(pb + 2), P(pb + 3),
        P(pb + 4), P(pb + 5), P(pb + 6), P(pb + 7), out, acc);
  };
  auto mlp128 = [&](const float* X, int N, int pb, float* out, int acc) {
    k_mlp<HID><<<dim3((N + 31) / 32), dim3(256), 0, stream>>>(
        X, N, P(pb), P(pb + 1), P(pb + 2), P(pb + 3),
        P(pb + 4), P(pb + 5), P(pb + 6), P(pb + 7), out, acc);
  };
  auto scatter = [&](const float* xs, const int* s, const int* d, float* agg, int E, bool wide) {
    if (wide) {
      long long total = (long long)E * 32;
      k_scatter128<<<dim3((int)((total + 255) / 256)), dim3(256), 0, stream>>>(xs, s, d, agg, E);
    } else {
      long long total = (long long)E * 8;
      k_scatter8<<<dim3((int)((total + 255) / 256)), dim3(256), 0, stream>>>(xs, s, d, agg, E);
    }
  };

  // layer-0 inputs + GIN self-term fan-out into the layer-0 agg buffers
  k_init_op_fan<<<dim3((NOP * HID + 255) / 256), dim3(256), 0, stream>>>(
      x_op, xop, aggA_op, aggB_op, NOP);
  k_init_m_fan <<<dim3((NM * HID + 255) / 256), dim3(256), 0, stream>>>(
      x_m, Wm_in, bm_in, xm, aggA_m, aggB_m, NM);

  for (int L = 0; L < 3; ++L) {
    const bool wide = (L != 0);
    // all 4 aggregations while xop/xm are live (agg buffers pre-seeded with x_dst)
    scatter(xop, src_oo, dst_oo, aggA_op, E_oo, wide);
    scatter(xm,  src_mo, dst_mo, aggB_op, E_mo, wide);
    scatter(xop, src_om, dst_om, aggA_m,  E_om, wide);
    scatter(xm,  src_mm, dst_mm, aggB_m,  E_mm, wide);
    const int base = 4 + L * 32;
    if (L == 0) {
      mlp7(aggA_op, NOP, base + 0,  out_op, 0);  // oo
      mlp7(aggB_op, NOP, base + 16, out_op, 1);  // mo (accumulate)
      mlp7(aggA_m,  NM,  base + 8,  out_m,  0);  // om
      mlp7(aggB_m,  NM,  base + 24, out_m,  1);  // mm (accumulate)
    } else {
      mlp128(aggA_op, NOP, base + 0,  out_op, 0);
      mlp128(aggB_op, NOP, base + 16, out_op, 1);
      mlp128(aggA_m,  NM,  base + 8,  out_m,  0);
      mlp128(aggB_m,  NM,  base + 24, out_m,  1);
    }
    if (L < 2) {
      // relu into next-layer features AND seed next layer's agg buffers (1 read, 3 writes)
      k_relu_fan<<<dim3((int)((opF + 255) / 256)), dim3(256), 0, stream>>>(
          out_op, xop, aggA_op, aggB_op, opF);
      k_relu_fan<<<dim3((int)((mF + 255) / 256)), dim3(256), 0, stream>>>(
          out_m, xm, aggA_m, aggB_m, mF);
    } else {
      k_relu_copy<<<dim3((int)((opF + 255) / 256)), dim3(256), 0, stream>>>(out_op, xop, opF);
      k_relu_copy<<<dim3((int)((mF  + 255) / 256)), dim3(256), 0, stream>>>(out_m,  xm,  mF);
    }
  }

  // pooled output: sum_i (x_i @ W + b) = colsum(x) @ W + N*b
  k_zero<<<dim3(1), dim3(256), 0, stream>>>(colsum, 256);
  k_colsum<<<dim3((NOP + 255) / 256), dim3(128), 0, stream>>>(xop, NOP, colsum);
  k_colsum<<<dim3((NM  + 255) / 256), dim3(128), 0, stream>>>(xm,  NM,  colsum + 128);
  k_final<<<dim3(1), dim3(32), 0, stream>>>(colsum, Wop, bop, Wm_out, bm_out, (float*)d_out);
}